// LocalCrossAttentionFuse_90245852824320
// MI455X (gfx1250) — compile-verified
//
#include <hip/hip_runtime.h>

typedef float v2f __attribute__((ext_vector_type(2)));
typedef float v8f __attribute__((ext_vector_type(8)));

__device__ __forceinline__ v8f wmma4(v2f a, v2f b, v8f c) {
  // V_WMMA_F32_16X16X4_F32 : D(16x16,f32) = A(16x4,f32) x B(4x16,f32) + C
  return __builtin_amdgcn_wmma_f32_16x16x4_f32(false, a, false, b, (short)0, c, false, false);
}

__device__ __forceinline__ v8f vzero8() {
  v8f z = {0.f, 0.f, 0.f, 0.f, 0.f, 0.f, 0.f, 0.f};
  return z;
}

// -------------------------------------------------------------------------
// 1) 4x4 average pool: [4,128,256,256] -> [4,128,64,64], both tensors.
// -------------------------------------------------------------------------
__global__ __launch_bounds__(256) void pool_kernel(const float* __restrict__ vv,
                                                   const float* __restrict__ vh,
                                                   float* __restrict__ vvp,
                                                   float* __restrict__ vhp) {
  int idx = blockIdx.x * 256 + threadIdx.x;     // 0 .. 2*2097152-1
  int sel = idx >> 21;
  int r = idx & 2097151;                        // (b*128+ch)*4096 + h4*64 + w4
  const float* src = sel ? vh : vv;
  float* dst = sel ? vhp : vvp;
  int w4 = r & 63;
  int h4 = (r >> 6) & 63;
  int plane = r >> 12;                          // b*128 + ch
  size_t base = ((size_t)plane * 256 + (size_t)h4 * 4) * 256 + (size_t)w4 * 4;
  float s = 0.f;
#pragma unroll
  for (int rr = 0; rr < 4; ++rr) {
    float4 v = *(const float4*)(src + base + rr * 256);
    s += v.x + v.y + v.z + v.w;
  }
  dst[r] = s * 0.0625f;
}

// -------------------------------------------------------------------------
// 2) Fold out_proj into final projection: Wo1 = P1@out_w, Wo2 = P2@out_w,
//    bo = (P1+P2)@out_b + proj_b.  proj_w: [128,256], out_w: [128,128].
// -------------------------------------------------------------------------
__global__ __launch_bounds__(256) void fold_kernel(const float* __restrict__ proj_w,
                                                   const float* __restrict__ out_w,
                                                   const float* __restrict__ out_b,
                                                   const float* __restrict__ proj_b,
                                                   float* __restrict__ WF,   // [2*16384]
                                                   float* __restrict__ BF) { // [128]
  int blk = blockIdx.x;
  int tid = threadIdx.x;
  if (blk < 128) {
    int mi = blk >> 6;                         // 0: Wo1, 1: Wo2
    int elem = (blk & 63) * 256 + tid;
    int o = elem >> 7, c = elem & 127;
    float acc = 0.f;
    for (int m = 0; m < 128; ++m)
      acc += proj_w[o * 256 + mi * 128 + m] * out_w[m * 128 + c];
    WF[mi * 16384 + o * 128 + c] = acc;
  } else if (tid < 128) {
    int o = tid;
    float acc = proj_b[o];
    for (int m = 0; m < 128; ++m)
      acc += (proj_w[o * 256 + m] + proj_w[o * 256 + 128 + m]) * out_b[m];
    BF[o] = acc;
  }
}

// -------------------------------------------------------------------------
// 3) Per-window fused: conv1x1 on pooled tile (WMMA) -> LDS, then in_proj
//    on the *scrambled* tokens (row-major (c,ws,ws)->(ws*ws,c) reshape):
//    X[t,c] = conv[ch = 2t + (c>>6), pos = c&63].
//    Output Y token-major [16384,128].
// -------------------------------------------------------------------------
__global__ __launch_bounds__(256) void qkv_kernel(const float* __restrict__ pooled, // [4,128,64,64]
                                                  const float* __restrict__ wconv,  // [128,128]
                                                  const float* __restrict__ bconv,  // [128]
                                                  const float* __restrict__ inw,    // [128,128] (slice)
                                                  const float* __restrict__ inb,    // [128] (slice)
                                                  float* __restrict__ Y) {          // [16384,128]
  __shared__ float Cs[64 * 128];  // conv tile: [local_pix 64][out_ch 128]
  int win = blockIdx.x;                         // 0..255
  int b = win >> 6, nh = (win >> 3) & 7, nw = win & 7;
  int spbase = (nh * 8) * 64 + nw * 8;
  int tid = threadIdx.x;
  int lane = tid & 31, wv = tid >> 5;
  int half = lane >> 4, lm = lane & 15;
  int mt = wv & 3;
  int m0 = mt * 16;

  // ---- GEMM-1: Cs[p, o] = sum_m pooled[b, m, pix(p)] * wconv[o, m] + bconv[o]
  {
    int p = m0 + lm;
    int sp = spbase + (p >> 3) * 64 + (p & 7);
    const float* arow = pooled + (size_t)b * 128 * 4096 + sp;
    for (int nt = (wv >> 2); nt < 8; nt += 2) {
      int n0 = nt * 16;
      const float* brow = wconv + (n0 + lm) * 128;
      v8f acc = vzero8();
      for (int kk = 0; kk < 128; kk += 4) {
        int k = kk + half * 2;
        v2f a;
        a.x = arow[(size_t)k * 4096];
        a.y = arow[(size_t)(k + 1) * 4096];
        float2 w2 = *(const float2*)(brow + k);
        v2f bb; bb.x = w2.x; bb.y = w2.y;
        acc = wmma4(a, bb, acc);
      }
      float bc = bconv[n0 + lm];
#pragma unroll
      for (int v = 0; v < 8; ++v)
        Cs[(m0 + v + half * 8) * 128 + n0 + lm] = acc[v] + bc;
    }
  }
  __syncthreads();

  // ---- GEMM-2: Y[t, o] = sum_c X[t,c] * inw[o,c] + inb[o]
  {
    int t = m0 + lm;
    for (int nt = (wv >> 2); nt < 8; nt += 2) {
      int n0 = nt * 16;
      const float* brow = inw + (n0 + lm) * 128;
      v8f acc = vzero8();
      for (int kk = 0; kk < 128; kk += 4) {
        int k = kk + half * 2;
        v2f a;
        a.x = Cs[(k & 63) * 128 + 2 * t + (k >> 6)];
        a.y = Cs[((k + 1) & 63) * 128 + 2 * t + ((k + 1) >> 6)];
        float2 w2 = *(const float2*)(brow + k);
        v2f bb; bb.x = w2.x; bb.y = w2.y;
        acc = wmma4(a, bb, acc);
      }
      float bi = inb[n0 + lm];
#pragma unroll
      for (int v = 0; v < 8; ++v)
        Y[(size_t)(win * 64 + m0 + v + half * 8) * 128 + n0 + lm] = acc[v] + bi;
    }
  }
}

// -------------------------------------------------------------------------
// 4) Attention: one wave per (window, head). S = (Q*scale)K^T (WMMA),
//    softmax rows in LDS, O = S@V (WMMA). L=64, d=32.
// -------------------------------------------------------------------------
__global__ __launch_bounds__(32) void attn_kernel(const float* __restrict__ Q,
                                                  const float* __restrict__ K,
                                                  const float* __restrict__ V,
                                                  float* __restrict__ O) {
  __shared__ float Qs[64 * 32];
  __shared__ float Ks[64 * 32];
  __shared__ float Vs[64 * 32];
  __shared__ float Ss[64 * 65];
  int win = blockIdx.x >> 2;
  int h = blockIdx.x & 3;
  int lane = threadIdx.x;
  int half = lane >> 4, lm = lane & 15;
  const float scale = 0.1767766952966369f;  // 1/sqrt(32)

  for (int i = lane; i < 2048; i += 32) {
    int r = i >> 5, c = i & 31;
    size_t g = (size_t)(win * 64 + r) * 128 + h * 32 + c;
    Qs[i] = Q[g] * scale;
    Ks[i] = K[g];
    Vs[i] = V[g];
  }
  __syncthreads();

  // scores S[l,m] = Q_h[l,:] . K_h[m,:]
  for (int mt = 0; mt < 4; ++mt)
    for (int nt = 0; nt < 4; ++nt) {
      v8f acc = vzero8();
      for (int kk = 0; kk < 32; kk += 4) {
        int k = kk + half * 2;
        v2f a; a.x = Qs[(mt * 16 + lm) * 32 + k]; a.y = Qs[(mt * 16 + lm) * 32 + k + 1];
        v2f bb; bb.x = Ks[(nt * 16 + lm) * 32 + k]; bb.y = Ks[(nt * 16 + lm) * 32 + k + 1];
        acc = wmma4(a, bb, acc);
      }
#pragma unroll
      for (int v = 0; v < 8; ++v)
        Ss[(mt * 16 + v + half * 8) * 65 + nt * 16 + lm] = acc[v];
    }
  __syncthreads();

  // softmax over columns, two rows per lane
  for (int rr = 0; rr < 2; ++rr) {
    int r = lane + rr * 32;
    float mx = -1e30f;
    for (int n = 0; n < 64; ++n) mx = fmaxf(mx, Ss[r * 65 + n]);
    float s = 0.f;
    for (int n = 0; n < 64; ++n) {
      float e = __expf(Ss[r * 65 + n] - mx);
      Ss[r * 65 + n] = e;
      s += e;
    }
    float inv = 1.f / s;
    for (int n = 0; n < 64; ++n) Ss[r * 65 + n] *= inv;
  }
  __syncthreads();

  // O = P @ V : [64,64] x [64,32]
  for (int mt = 0; mt < 4; ++mt)
    for (int nt = 0; nt < 2; ++nt) {
      v8f acc = vzero8();
      for (int kk = 0; kk < 64; kk += 4) {
        int k = kk + half * 2;
        v2f a; a.x = Ss[(mt * 16 + lm) * 65 + k]; a.y = Ss[(mt * 16 + lm) * 65 + k + 1];
        v2f bb; bb.x = Vs[k * 32 + nt * 16 + lm]; bb.y = Vs[(k + 1) * 32 + nt * 16 + lm];
        acc = wmma4(a, bb, acc);
      }
#pragma unroll
      for (int v = 0; v < 8; ++v)
        O[(size_t)(win * 64 + mt * 16 + v + half * 8) * 128 + h * 32 + nt * 16 + lm] = acc[v];
    }
}

// -------------------------------------------------------------------------
// 5) Folded projection: olow[b,c,ys,xs] = O1@Wo1^T + O2@Wo2^T + bo
//    (window_unpart is the *natural* layout: token row t -> (y=t/8, x=t%8)).
// -------------------------------------------------------------------------
__global__ __launch_bounds__(256) void proj_kernel(const float* __restrict__ O1,
                                                   const float* __restrict__ O2,
                                                   const float* __restrict__ Wo1,
                                                   const float* __restrict__ Wo2,
                                                   const float* __restrict__ bo,
                                                   float* __restrict__ olow) { // [4,128,64,64]
  int win = blockIdx.x;
  int b = win >> 6, nh = (win >> 3) & 7, nw = win & 7;
  int tid = threadIdx.x;
  int lane = tid & 31, wv = tid >> 5;
  int half = lane >> 4, lm = lane & 15;
  int m0 = (wv & 3) * 16;
  const float* a1row = O1 + (size_t)(win * 64 + m0 + lm) * 128;
  const float* a2row = O2 + (size_t)(win * 64 + m0 + lm) * 128;
  for (int nt = (wv >> 2); nt < 8; nt += 2) {
    int n0 = nt * 16;
    const float* w1row = Wo1 + (n0 + lm) * 128;
    const float* w2row = Wo2 + (n0 + lm) * 128;
    v8f acc = vzero8();
    for (int kk = 0; kk < 128; kk += 4) {
      int k = kk + half * 2;
      float2 t2;
      v2f a, bb;
      t2 = *(const float2*)(a1row + k); a.x = t2.x; a.y = t2.y;
      t2 = *(const float2*)(w1row + k); bb.x = t2.x; bb.y = t2.y;
      acc = wmma4(a, bb, acc);
      t2 = *(const float2*)(a2row + k); a.x = t2.x; a.y = t2.y;
      t2 = *(const float2*)(w2row + k); bb.x = t2.x; bb.y = t2.y;
      acc = wmma4(a, bb, acc);
    }
    float bb2 = bo[n0 + lm];
#pragma unroll
    for (int v = 0; v < 8; ++v) {
      int m = m0 + v + half * 8;  // token -> (y,x)
      olow[((size_t)(b * 128 + n0 + lm) * 64 + nh * 8 + (m >> 3)) * 64 + nw * 8 + (m & 7)] =
          acc[v] + bb2;
    }
  }
}

// -------------------------------------------------------------------------
// 6) Bilinear upsample x4 (half-pixel centers, clamp) -> d_out [4,128,256,256]
// -------------------------------------------------------------------------
__global__ __launch_bounds__(256) void upsample_kernel(const float* __restrict__ low,
                                                       float* __restrict__ out) {
  int idx = blockIdx.x * 256 + threadIdx.x;
  int j = idx & 255;
  int i = (idx >> 8) & 255;
  int plane = idx >> 16;  // b*128 + ch
  float fy = (i + 0.5f) * 0.25f - 0.5f;
  float fx = (j + 0.5f) * 0.25f - 0.5f;
  float y0f = floorf(fy), x0f = floorf(fx);
  float dy = fy - y0f, dx = fx - x0f;
  int y0 = (int)y0f, x0 = (int)x0f;
  int y1 = min(63, max(0, y0 + 1)), x1 = min(63, max(0, x0 + 1));
  y0 = min(63, max(0, y0));
  x0 = min(63, max(0, x0));
  const float* p = low + (size_t)plane * 4096;
  float v00 = p[y0 * 64 + x0], v01 = p[y0 * 64 + x1];
  float v10 = p[y1 * 64 + x0], v11 = p[y1 * 64 + x1];
  out[idx] = (1.f - dy) * ((1.f - dx) * v00 + dx * v01) +
             dy * ((1.f - dx) * v10 + dx * v11);
}

// -------------------------------------------------------------------------
extern "C" void kernel_launch(void* const* d_in, const int* in_sizes, int n_in,
                              void* d_out, int out_size, void* d_ws, size_t ws_size,
                              hipStream_t stream) {
  (void)in_sizes; (void)n_in; (void)out_size; (void)ws_size;
  const float* vv = (const float*)d_in[0];
  const float* vh = (const float*)d_in[1];
  const float* w_q_vv = (const float*)d_in[2];  const float* b_q_vv = (const float*)d_in[3];
  const float* w_k_vh = (const float*)d_in[4];  const float* b_k_vh = (const float*)d_in[5];
  const float* w_v_vh = (const float*)d_in[6];  const float* b_v_vh = (const float*)d_in[7];
  const float* w_q_vh = (const float*)d_in[8];  const float* b_q_vh = (const float*)d_in[9];
  const float* w_k_vv = (const float*)d_in[10]; const float* b_k_vv = (const float*)d_in[11];
  const float* w_v_vv = (const float*)d_in[12]; const float* b_v_vv = (const float*)d_in[13];
  const float* in_w = (const float*)d_in[14];   const float* in_b = (const float*)d_in[15];
  const float* out_w = (const float*)d_in[16];  const float* out_b = (const float*)d_in[17];
  const float* proj_w = (const float*)d_in[18]; const float* proj_b = (const float*)d_in[19];
  float* out = (float*)d_out;

  const size_t M = 2097152;  // 4*128*64*64
  float* W = (float*)d_ws;
  float* VVP = W;
  float* VHP = W + 1 * M;
  float* Q1  = W + 2 * M;
  float* K1  = W + 3 * M;
  float* V1  = W + 4 * M;
  float* Q2  = W + 5 * M;
  float* K2  = W + 6 * M;
  float* V2  = W + 7 * M;
  float* O1  = W + 8 * M;
  float* O2  = W + 9 * M;
  float* OLOW = W + 10 * M;
  float* WF  = W + 11 * M;          // 2*16384
  float* BF  = WF + 2 * 16384;      // 128

  pool_kernel<<<16384, 256, 0, stream>>>(vv, vh, VVP, VHP);
  fold_kernel<<<129, 256, 0, stream>>>(proj_w, out_w, out_b, proj_b, WF, BF);

  // attend 1: q from vv, k/v from vh
  qkv_kernel<<<256, 256, 0, stream>>>(VVP, w_q_vv, b_q_vv, in_w + 0 * 16384, in_b + 0,   Q1);
  qkv_kernel<<<256, 256, 0, stream>>>(VHP, w_k_vh, b_k_vh, in_w + 1 * 16384, in_b + 128, K1);
  qkv_kernel<<<256, 256, 0, stream>>>(VHP, w_v_vh, b_v_vh, in_w + 2 * 16384, in_b + 256, V1);
  // attend 2: q from vh, k/v from vv
  qkv_kernel<<<256, 256, 0, stream>>>(VHP, w_q_vh, b_q_vh, in_w + 0 * 16384, in_b + 0,   Q2);
  qkv_kernel<<<256, 256, 0, stream>>>(VVP, w_k_vv, b_k_vv, in_w + 1 * 16384, in_b + 128, K2);
  qkv_kernel<<<256, 256, 0, stream>>>(VVP, w_v_vv, b_v_vv, in_w + 2 * 16384, in_b + 256, V2);

  attn_kernel<<<1024, 32, 0, stream>>>(Q1, K1, V1, O1);
  attn_kernel<<<1024, 32, 0, stream>>>(Q2, K2, V2, O2);

  proj_kernel<<<256, 256, 0, stream>>>(O1, O2, WF, WF + 16384, BF, OLOW);
  upsample_kernel<<<131072, 256, 0, stream>>>(OLOW, out);
}